// ProteinStateReconstructor2D_8641474199775
// MI455X (gfx1250) — compile-verified
//
#include <hip/hip_runtime.h>

#define BATCH 8
#define NATOM 20000
#define DCH   640
#define RSEG  2500
#define BINS  10
#define PSTR  16                    // padded pooled row stride (floats)
#define HID   128
#define NROWS (BATCH * NATOM)       // 160000
#define NTILES (NROWS / 16)         // 10000

typedef __attribute__((ext_vector_type(2))) float v2f;
typedef __attribute__((ext_vector_type(4))) float v4f;
typedef __attribute__((ext_vector_type(8))) float v8f;

// ---------------------------------------------------------------------------
// Kernel 1: segment mean + bin pooling. One block per (b, residue).
// 160 threads = 5 waves; thread t owns channels [4t, 4t+3] (one b128 load per
// atom, coalesced 2560B rows). Each 16-lane half-wave owns exactly one bin
// (bin = t>>4): reduction is 4 shfl_xor steps + one direct store.
// x is streamed once (410MB > 192MB L2) -> non-temporal loads.
// ---------------------------------------------------------------------------
__device__ __forceinline__ int lower_bound_i32(const int* a, int n, int key) {
    int lo = 0, hi = n;
    while (lo < hi) {
        int mid = (lo + hi) >> 1;
        if (a[mid] < key) lo = mid + 1; else hi = mid;
    }
    return lo;
}

__global__ __launch_bounds__(160) void seg_pool_kernel(
    const float* __restrict__ x, const int* __restrict__ res_id,
    float* __restrict__ pooled)
{
    const int bid = blockIdx.x;
    const int b = bid / RSEG;
    const int r = bid - b * RSEG;
    const int t = threadIdx.x;              // 0..159

    const int start = lower_bound_i32(res_id, NATOM, r);
    const int end   = lower_bound_i32(res_id, NATOM, r + 1);
    const int count = end - start;

    // float4 view: row of 640 floats = 160 float4s; thread t takes the t-th.
    const v4f* base = (const v4f*)x + (size_t)(b * NATOM + start) * 160 + t;

    v4f acc = {0.f, 0.f, 0.f, 0.f};
    for (int a = 0; a < count; ++a) {
        v4f v = __builtin_nontemporal_load(base + (size_t)a * 160);
        acc += v;
    }
    float s = acc.x + acc.y + acc.z + acc.w;
    // Sum across the 16-lane group that shares this bin.
    s += __shfl_xor(s, 1, 32);
    s += __shfl_xor(s, 2, 32);
    s += __shfl_xor(s, 4, 32);
    s += __shfl_xor(s, 8, 32);

    if ((t & 15) == 0) {
        const int bin = t >> 4;             // 0..9
        const float scale = (count > 0) ? 1.0f / (64.0f * (float)count) : 0.0f;
        pooled[(size_t)(b * RSEG + r) * PSTR + bin] = s * scale;
    }
}

// ---------------------------------------------------------------------------
// Shared WMMA helper: C(16x16) = sum_kk A(16x4) * B(4x16), f32, K padded to 16
// ---------------------------------------------------------------------------
__device__ __forceinline__ v8f mma_tile(const v2f a[4], const v2f bfr[4]) {
    v8f c = {0.f, 0.f, 0.f, 0.f, 0.f, 0.f, 0.f, 0.f};
#pragma unroll
    for (int kk = 0; kk < 4; ++kk) {
        c = __builtin_amdgcn_wmma_f32_16x16x4_f32(
            /*neg_a=*/false, a[kk], /*neg_b=*/false, bfr[kk],
            /*c_mod=*/(short)0, c, /*reuse_a=*/false, /*reuse_b=*/false);
    }
    return c;
}

// Load features for one row directly into A-fragments. Named scalars /
// vector lanes only (no addressable array -> no dynamic-index select chains).
// Lanes 0-15 take K={4kk,4kk+1}; lanes 16-31 take K={4kk+2,4kk+3}.
// Feature k: k<10 -> pooled bin k; 10..12 -> conditioner; 13..15 -> 0.
__device__ __forceinline__ void load_af(
    int row, const float* __restrict__ pooled, const float* __restrict__ cond,
    const int* __restrict__ res_id, bool lo16, v2f Af[4])
{
    const int b = row / NATOM;
    const int n = row - b * NATOM;
    const int r = res_id[n];
    const float* pp = pooled + ((size_t)b * RSEG + r) * PSTR;
    const v4f q0 = *(const v4f*)(pp);       // f0..f3   (64B aligned)
    const v4f q1 = *(const v4f*)(pp + 4);   // f4..f7
    const v2f q2 = *(const v2f*)(pp + 8);   // f8,f9
    const float f10 = cond[n * 3 + 0];
    const float f11 = cond[n * 3 + 1];
    const float f12 = cond[n * 3 + 2];

    Af[0].x = lo16 ? q0.x : q0.z;   // K0 | K2
    Af[0].y = lo16 ? q0.y : q0.w;   // K1 | K3
    Af[1].x = lo16 ? q1.x : q1.z;   // K4 | K6
    Af[1].y = lo16 ? q1.y : q1.w;   // K5 | K7
    Af[2].x = lo16 ? q2.x : f10;    // K8 | K10
    Af[2].y = lo16 ? q2.y : f11;    // K9 | K11
    Af[3].x = lo16 ? f12 : 0.f;     // K12| K14(=0)
    Af[3].y = 0.f;                  // K13/K15 = 0
}

// ---------------------------------------------------------------------------
// Kernel 2: batch-norm statistics. h = relu(feat@W1+b1), accumulate sum/sumsq.
// ---------------------------------------------------------------------------
__global__ __launch_bounds__(256) void stats_kernel(
    const float* __restrict__ pooled, const float* __restrict__ cond,
    const int* __restrict__ res_id, const float* __restrict__ W1,
    const float* __restrict__ b1, float* __restrict__ gSum,
    float* __restrict__ gSq)
{
    __shared__ float W1s[16 * HID];
    __shared__ float b1s[HID];
    __shared__ float sSum[HID];
    __shared__ float sSq[HID];

    const int tid = threadIdx.x;
    for (int i = tid; i < 16 * HID; i += 256) {
        int k = i >> 7, c = i & 127;
        W1s[i] = (k < 13) ? W1[k * HID + c] : 0.0f;
    }
    if (tid < HID) { b1s[tid] = b1[tid]; sSum[tid] = 0.f; sSq[tid] = 0.f; }
    __syncthreads();

    const int lane = tid & 31;
    const int wave = tid >> 5;
    const int lmod = lane & 15;
    const bool lo16 = (lane < 16);
    const int koff = lo16 ? 0 : 2;

    // Hoist B fragments (W1) into registers: loop-invariant.
    v2f Bf[8][4];
#pragma unroll
    for (int j = 0; j < 8; ++j)
#pragma unroll
        for (int kk = 0; kk < 4; ++kk) {
            int k0 = 4 * kk + koff;
            Bf[j][kk].x = W1s[k0 * HID + 16 * j + lmod];
            Bf[j][kk].y = W1s[(k0 + 1) * HID + 16 * j + lmod];
        }

    const int stride = gridDim.x * 8;
    for (int tile = blockIdx.x * 8 + wave; tile < NTILES; tile += stride) {
        const int row = tile * 16 + lmod;
        v2f Af[4];
        load_af(row, pooled, cond, res_id, lo16, Af);
#pragma unroll
        for (int j = 0; j < 8; ++j) {
            v8f cacc = mma_tile(Af, Bf[j]);
            const int col = 16 * j + lmod;
            const float bb = b1s[col];
            float s = 0.f, q = 0.f;
#pragma unroll
            for (int i = 0; i < 8; ++i) {
                float h = cacc[i] + bb;
                h = h > 0.f ? h : 0.f;
                s += h; q += h * h;
            }
            atomicAdd(&sSum[col], s);
            atomicAdd(&sSq[col], q);
        }
    }
    __syncthreads();
    if (tid < HID) {
        atomicAdd(&gSum[tid], sSum[tid]);
        atomicAdd(&gSq[tid], sSq[tid]);
    }
}

// ---------------------------------------------------------------------------
// Kernel 3: fold mean/var + gamma/beta into scale/shift per column.
// ---------------------------------------------------------------------------
__global__ void finalize_kernel(
    const float* __restrict__ gSum, const float* __restrict__ gSq,
    const float* __restrict__ gamma, const float* __restrict__ beta,
    float* __restrict__ scaleArr, float* __restrict__ shiftArr)
{
    const int c = threadIdx.x;
    if (c < HID) {
        const float invM = 1.0f / (float)NROWS;
        float mu  = gSum[c] * invM;
        float var = gSq[c] * invM - mu * mu;
        var = var > 0.f ? var : 0.f;
        float rstd = rsqrtf(var + 1e-5f);
        float sc = rstd * gamma[c];
        scaleArr[c] = sc;
        shiftArr[c] = beta[c] - mu * sc;
    }
}

// ---------------------------------------------------------------------------
// Kernel 4: recompute h via WMMA, normalize, apply W2 (128->3) via lane
// butterfly reduction, write output.
// ---------------------------------------------------------------------------
__global__ __launch_bounds__(256) void out_kernel(
    const float* __restrict__ pooled, const float* __restrict__ cond,
    const int* __restrict__ res_id, const float* __restrict__ W1,
    const float* __restrict__ b1, const float* __restrict__ scaleArr,
    const float* __restrict__ shiftArr, const float* __restrict__ W2,
    const float* __restrict__ b2, float* __restrict__ out)
{
    __shared__ float W1s[16 * HID];
    __shared__ float b1s[HID], scs[HID], shs[HID];
    __shared__ float W2s[HID * 3];
    __shared__ float b2s[3];

    const int tid = threadIdx.x;
    for (int i = tid; i < 16 * HID; i += 256) {
        int k = i >> 7, c = i & 127;
        W1s[i] = (k < 13) ? W1[k * HID + c] : 0.0f;
    }
    if (tid < HID) {
        b1s[tid] = b1[tid];
        scs[tid] = scaleArr[tid];
        shs[tid] = shiftArr[tid];
    }
    for (int i = tid; i < HID * 3; i += 256) W2s[i] = W2[i];
    if (tid < 3) b2s[tid] = b2[tid];
    __syncthreads();

    const int lane = tid & 31;
    const int wave = tid >> 5;
    const int lmod = lane & 15;
    const bool lo16 = (lane < 16);
    const int koff = lo16 ? 0 : 2;

    v2f Bf[8][4];
#pragma unroll
    for (int j = 0; j < 8; ++j)
#pragma unroll
        for (int kk = 0; kk < 4; ++kk) {
            int k0 = 4 * kk + koff;
            Bf[j][kk].x = W1s[k0 * HID + 16 * j + lmod];
            Bf[j][kk].y = W1s[(k0 + 1) * HID + 16 * j + lmod];
        }

    const int stride = gridDim.x * 8;
    for (int tile = blockIdx.x * 8 + wave; tile < NTILES; tile += stride) {
        const int row = tile * 16 + lmod;
        v2f Af[4];
        load_af(row, pooled, cond, res_id, lo16, Af);

        // acc[i][jj]: partial out for row (i or i+8 depending on half-wave)
        float acc[8][3];
#pragma unroll
        for (int i = 0; i < 8; ++i) { acc[i][0] = 0.f; acc[i][1] = 0.f; acc[i][2] = 0.f; }

#pragma unroll
        for (int j = 0; j < 8; ++j) {
            v8f cacc = mma_tile(Af, Bf[j]);
            const int col = 16 * j + lmod;
            const float bb = b1s[col], sc = scs[col], sh = shs[col];
            const float w20 = W2s[col * 3 + 0];
            const float w21 = W2s[col * 3 + 1];
            const float w22 = W2s[col * 3 + 2];
#pragma unroll
            for (int i = 0; i < 8; ++i) {
                float h = cacc[i] + bb;
                h = h > 0.f ? h : 0.f;
                float hn = h * sc + sh;
                acc[i][0] += hn * w20;
                acc[i][1] += hn * w21;
                acc[i][2] += hn * w22;
            }
        }
        // Butterfly-sum across the 16 lanes of each half-wave (column groups).
#pragma unroll
        for (int i = 0; i < 8; ++i)
#pragma unroll
            for (int jj = 0; jj < 3; ++jj) {
#pragma unroll
                for (int mask = 1; mask < 16; mask <<= 1)
                    acc[i][jj] += __shfl_xor(acc[i][jj], mask, 32);
            }
        // Lanes 0-15 hold rows tile*16+0..7 (i=0..7); lanes 16-31 rows +8..15.
        const int m0 = tile * 16 + (lo16 ? 0 : 8);
#pragma unroll
        for (int i = 0; i < 8; ++i) {
            if (lmod == i) {
                const int rrow = m0 + i;
                out[rrow * 3 + 0] = acc[i][0] + b2s[0];
                out[rrow * 3 + 1] = acc[i][1] + b2s[1];
                out[rrow * 3 + 2] = acc[i][2] + b2s[2];
            }
        }
    }
}

// ---------------------------------------------------------------------------
extern "C" void kernel_launch(void* const* d_in, const int* in_sizes, int n_in,
                              void* d_out, int out_size, void* d_ws, size_t ws_size,
                              hipStream_t stream) {
    const float* x      = (const float*)d_in[0];
    const float* cond   = (const float*)d_in[1];
    const int*   res_id = (const int*)d_in[2];
    const float* W1     = (const float*)d_in[3];
    const float* b1     = (const float*)d_in[4];
    const float* gamma  = (const float*)d_in[5];
    const float* beta   = (const float*)d_in[6];
    const float* W2     = (const float*)d_in[7];
    const float* b2     = (const float*)d_in[8];
    float* out = (float*)d_out;

    float* ws      = (float*)d_ws;
    float* pooled  = ws;                       // B*R*PSTR = 320000 floats
    float* gSum    = ws + (BATCH * RSEG * PSTR);
    float* gSq     = gSum + HID;               // 128
    float* scaleA  = gSq + HID;                // 128
    float* shiftA  = scaleA + HID;             // 128

    hipMemsetAsync(gSum, 0, 2 * HID * sizeof(float), stream);

    seg_pool_kernel<<<BATCH * RSEG, 160, 0, stream>>>(x, res_id, pooled);
    stats_kernel<<<625, 256, 0, stream>>>(pooled, cond, res_id, W1, b1, gSum, gSq);
    finalize_kernel<<<1, 128, 0, stream>>>(gSum, gSq, gamma, beta, scaleA, shiftA);
    out_kernel<<<625, 256, 0, stream>>>(pooled, cond, res_id, W1, b1,
                                        scaleA, shiftA, W2, b2, out);
}